// LlamaAttention_GEAR_48052094107667
// MI455X (gfx1250) — compile-verified
//
#include <hip/hip_runtime.h>
#include <math.h>

#define B_    4
#define H_    32
#define D_    128
#define HID_  4096
#define SQ_   4096
#define SF_   63
#define GS_   64
#define RANK_ 4
#define NKV_  (SQ_ + SF_ + 1)   // 4160
#define NG_   (SQ_ / GS_)       // 64
#define MPAD_ 16                // padded M (batch) for WMMA

typedef __attribute__((ext_vector_type(2))) float v2f;
typedef __attribute__((ext_vector_type(8))) float v8f;

// ---------------------------------------------------------------------------
// Build zero-padded (16, HID) input for the WMMA GEMMs:
//  - xpad rows 0..3 = hidden, rows 4..15 = 0
//  - apad rows 4..15 = 0 (rows 0..3 written later by attn_kernel)
// ---------------------------------------------------------------------------
__global__ __launch_bounds__(256) void init_pad_kernel(
    const float* __restrict__ hidden,
    float* __restrict__ xpad, float* __restrict__ apad)
{
    int i = blockIdx.x * blockDim.x + threadIdx.x;   // 0 .. 16*HID-1
    xpad[i] = (i < B_ * HID_) ? hidden[i] : 0.0f;
    if (i >= B_ * HID_) apad[i] = 0.0f;
}

// ---------------------------------------------------------------------------
// C = X @ W^T via V_WMMA_F32_16X16X4_F32, X is (16, HID) zero-padded.
// One wave per 16-channel tile.  Branch-free inner loop (EXEC stays all-1s).
// ---------------------------------------------------------------------------
__global__ __launch_bounds__(256) void gemm16_kernel(
    const float* __restrict__ X,   // (16, HID_), rows >= B_ are zero
    const float* __restrict__ W,   // (HID_, HID_)
    float* __restrict__ OUT)       // (B_, HID_)
{
    int wave = (blockIdx.x * blockDim.x + threadIdx.x) >> 5;  // 0..255
    int lane = threadIdx.x & 31;
    int m0   = wave * 16;
    int n     = lane & 15;          // B/C column (padded batch row)
    int khalf = (lane >> 4) * 2;    // K sub-offset for this lane half

    const float* wrow = W + (size_t)(m0 + n) * HID_ + khalf;  // A row M = m0+n
    const float* xrow = X + (size_t)n * HID_ + khalf;         // B col n

    v8f c = {};
    #pragma unroll 4
    for (int k = 0; k < HID_; k += 4) {
        v2f a, b;
        a.x = wrow[k];  a.y = wrow[k + 1];
        b.x = xrow[k];  b.y = xrow[k + 1];
        c = __builtin_amdgcn_wmma_f32_16x16x4_f32(false, a, false, b,
                                                  (short)0, c, false, false);
    }

    // C 16x16 f32: VGPR r -> M = r + (lane>=16 ? 8 : 0), N = lane&15
    if (n < B_) {
        int mo = (lane >> 4) * 8;
        #pragma unroll
        for (int r = 0; r < 8; ++r)
            OUT[(size_t)n * HID_ + m0 + mo + r] = c[r];
    }
}

// ---------------------------------------------------------------------------
// In-place RoPE on q_ws and k_ws (layout (B_, H_*D_)).
// ---------------------------------------------------------------------------
__global__ __launch_bounds__(256) void rope_kernel(
    float* __restrict__ q_ws, float* __restrict__ k_ws,
    const int* __restrict__ position_ids)
{
    int g = blockIdx.x * blockDim.x + threadIdx.x;
    float* buf = (g < B_ * HID_) ? q_ws : k_ws;
    int idx = g & (B_ * HID_ - 1);
    int b = idx / HID_;
    int d = idx & (D_ - 1);

    float pos = (float)position_ids[b];                  // (B, QL=1)
    int fi = d & 63;
    float inv_freq = expf(-((float)(2 * fi) / (float)D_) * logf(10000.0f));
    float ang = pos * inv_freq;
    float cv = cosf(ang), sv = sinf(ang);

    int pidx = idx + ((d < 64) ? 64 : -64);              // same 128-aligned head
    float x  = buf[idx];
    float xr = buf[pidx];
    float rot = (d < 64) ? -xr : xr;
    __syncthreads();                                     // all reads before writes
    buf[idx] = x * cv + rot * sv;
}

// ---------------------------------------------------------------------------
// One block per (b,h).  256 threads.
// ---------------------------------------------------------------------------
__global__ __launch_bounds__(256) void attn_kernel(
    const float* __restrict__ q_ws, const float* __restrict__ k_ws,
    const float* __restrict__ v_ws,
    const int*   __restrict__ k_quant, const float* __restrict__ k_scale,
    const float* __restrict__ k_mn,    const float* __restrict__ k_full,
    const float* __restrict__ key_p,   const float* __restrict__ key_q,
    const int*   __restrict__ v_quant, const float* __restrict__ v_scale,
    const float* __restrict__ v_mn,    const float* __restrict__ v_full,
    const float* __restrict__ value_q, const float* __restrict__ value_p,
    float* __restrict__ attn_out)      // padded (16, HID_) buffer, rows 0..3
{
    __shared__ float q_s[D_];
    __shared__ float coef[D_ * NG_];     // coef[d*NG_+g] = q[d]*k_scale[d][g]
    __shared__ float mn_dot[NG_];
    __shared__ float probs[NKV_];
    __shared__ float red[256];
    __shared__ float out_s[D_];
    __shared__ float ar_s[RANK_];

    const int bh = blockIdx.x;
    const int b = bh / H_, h = bh % H_;
    const int t = threadIdx.x;
    const float inv_sqrt_d = 0.08838834764831845f;       // 1/sqrt(128)

    const float* qv = q_ws + (size_t)b * HID_ + h * D_;
    if (t < D_) q_s[t] = qv[t];
    __syncthreads();

    // ---- dequant coefficients in LDS -------------------------------------
    const float* ks = k_scale + (size_t)bh * D_ * NG_;
    const float* km = k_mn    + (size_t)bh * D_ * NG_;
    for (int i = t; i < D_ * NG_; i += 256)
        coef[i] = q_s[i >> 6] * ks[i];
    if (t < NG_) {
        float acc = 0.0f;
        for (int d = 0; d < D_; ++d) acc += q_s[d] * km[d * NG_ + t];
        mn_dot[t] = acc;
    }
    __syncthreads();

    // ---- low-rank q projection (redundant per thread; cheap) -------------
    const float* kq = key_q + (size_t)bh * D_ * RANK_;
    float qr0 = 0, qr1 = 0, qr2 = 0, qr3 = 0;
    for (int d = 0; d < D_; ++d) {
        float qd = q_s[d];
        qr0 += qd * kq[d * 4 + 0]; qr1 += qd * kq[d * 4 + 1];
        qr2 += qd * kq[d * 4 + 2]; qr3 += qd * kq[d * 4 + 3];
    }

    // ---- quantized scores: 4 consecutive s per thread, b128 loads --------
    const int*   kqnt = k_quant + (size_t)bh * D_ * SQ_;
    const float* kp   = key_p   + (size_t)bh * SQ_ * RANK_;
    for (int i = 0; i < SQ_ / (256 * 4); ++i) {          // 4 iterations
        int s0 = i * 1024 + t * 4;
        int g = s0 >> 6;                                  // same group for s0..s0+3
        float base = mn_dot[g];
        float a0 = base, a1 = base, a2 = base, a3 = base;
        #pragma unroll 4
        for (int d = 0; d < D_; ++d) {
            const int4 cd = *(const int4*)(kqnt + (size_t)d * SQ_ + s0);
            float c = coef[d * NG_ + g];
            a0 += c * (float)cd.x; a1 += c * (float)cd.y;
            a2 += c * (float)cd.z; a3 += c * (float)cd.w;
        }
        const float4* kp4 = (const float4*)(kp + (size_t)s0 * 4);
        float4 p0 = kp4[0], p1 = kp4[1], p2 = kp4[2], p3 = kp4[3];
        a0 += qr0 * p0.x + qr1 * p0.y + qr2 * p0.z + qr3 * p0.w;
        a1 += qr0 * p1.x + qr1 * p1.y + qr2 * p1.z + qr3 * p1.w;
        a2 += qr0 * p2.x + qr1 * p2.y + qr2 * p2.z + qr3 * p2.w;
        a3 += qr0 * p3.x + qr1 * p3.y + qr2 * p3.z + qr3 * p3.w;
        probs[s0 + 0] = a0 * inv_sqrt_d;
        probs[s0 + 1] = a1 * inv_sqrt_d;
        probs[s0 + 2] = a2 * inv_sqrt_d;
        probs[s0 + 3] = a3 * inv_sqrt_d;
    }

    // ---- full-precision scores (63 cached rows + current k) --------------
    const float* kf   = k_full + (size_t)bh * SF_ * D_;
    const float* kcur = k_ws + (size_t)b * HID_ + h * D_;
    if (t < SF_ + 1) {
        const float* row = (t < SF_) ? (kf + (size_t)t * D_) : kcur;
        float acc = 0.0f;
        for (int d = 0; d < D_; ++d) acc += q_s[d] * row[d];
        probs[SQ_ + t] = acc * inv_sqrt_d;
    }
    __syncthreads();

    // ---- softmax ---------------------------------------------------------
    float lmax = -1e30f;
    for (int s = t; s < NKV_; s += 256) lmax = fmaxf(lmax, probs[s]);
    red[t] = lmax; __syncthreads();
    if (t == 0) { float m = red[0]; for (int i = 1; i < 256; ++i) m = fmaxf(m, red[i]); red[0] = m; }
    __syncthreads();
    float gmax = red[0];
    __syncthreads();
    float lsum = 0.0f;
    for (int s = t; s < NKV_; s += 256) { float e = expf(probs[s] - gmax); probs[s] = e; lsum += e; }
    red[t] = lsum; __syncthreads();
    if (t == 0) { float x = 0; for (int i = 0; i < 256; ++i) x += red[i]; red[0] = 1.0f / x; }
    __syncthreads();
    float rinv = red[0];
    for (int s = t; s < NKV_; s += 256) probs[s] *= rinv;
    if (t < D_) out_s[t] = 0.0f;
    __syncthreads();

    // ---- out = aw_q @ v_dq: 4 channels/lane, 8 s-phases, b128 loads ------
    const int*   vqnt = v_quant + (size_t)bh * SQ_ * D_;
    const float* vs   = v_scale + (size_t)bh * SQ_ * 2;
    const float* vm   = v_mn    + (size_t)bh * SQ_ * 2;
    {
        int d4    = (t & 31) * 4;      // channel base (0..124)
        int phase = t >> 5;            // 0..7
        int gd    = d4 >> 6;           // half-wave uniform
        float o0 = 0, o1 = 0, o2 = 0, o3 = 0;
        for (int s = phase; s < SQ_; s += 8) {
            float aw = probs[s];
            const int4 cd = *(const int4*)(vqnt + (size_t)s * D_ + d4);
            float sc = vs[s * 2 + gd], mn = vm[s * 2 + gd];
            o0 += aw * ((float)cd.x * sc + mn);
            o1 += aw * ((float)cd.y * sc + mn);
            o2 += aw * ((float)cd.z * sc + mn);
            o3 += aw * ((float)cd.w * sc + mn);
        }
        atomicAdd(&out_s[d4 + 0], o0);
        atomicAdd(&out_s[d4 + 1], o1);
        atomicAdd(&out_s[d4 + 2], o2);
        atomicAdd(&out_s[d4 + 3], o3);
    }
    __syncthreads();

    // ---- low-rank value correction ---------------------------------------
    const float* vq = value_q + (size_t)bh * SQ_ * RANK_;
    float a0 = 0, a1 = 0, a2 = 0, a3 = 0;
    for (int s = t; s < SQ_; s += 256) {
        float aw = probs[s];
        float4 v4 = *(const float4*)(vq + (size_t)s * 4);
        a0 += aw * v4.x; a1 += aw * v4.y; a2 += aw * v4.z; a3 += aw * v4.w;
    }
    for (int r = 0; r < RANK_; ++r) {
        float v = (r == 0) ? a0 : (r == 1) ? a1 : (r == 2) ? a2 : a3;
        red[t] = v; __syncthreads();
        if (t == 0) { float x = 0; for (int i = 0; i < 256; ++i) x += red[i]; ar_s[r] = x; }
        __syncthreads();
    }

    // ---- full-V part + write ---------------------------------------------
    const float* vf   = v_full + (size_t)bh * SF_ * D_;
    const float* vcur = v_ws + (size_t)b * HID_ + h * D_;
    const float* vp   = value_p + (size_t)bh * D_ * RANK_;
    if (t < D_) {
        float o = out_s[t];
        float4 vp4 = *(const float4*)(vp + (size_t)t * 4);
        o += ar_s[0] * vp4.x + ar_s[1] * vp4.y + ar_s[2] * vp4.z + ar_s[3] * vp4.w;
        for (int s2 = 0; s2 < SF_; ++s2) o += probs[SQ_ + s2] * vf[(size_t)s2 * D_ + t];
        o += probs[SQ_ + SF_] * vcur[t];
        attn_out[(size_t)b * HID_ + h * D_ + t] = o;
    }
}

// ---------------------------------------------------------------------------
extern "C" void kernel_launch(void* const* d_in, const int* in_sizes, int n_in,
                              void* d_out, int out_size, void* d_ws, size_t ws_size,
                              hipStream_t stream)
{
    (void)in_sizes; (void)n_in; (void)out_size; (void)ws_size;

    const float* hidden  = (const float*)d_in[0];
    const float* wq      = (const float*)d_in[1];
    const float* wk      = (const float*)d_in[2];
    const float* wv      = (const float*)d_in[3];
    const float* wo      = (const float*)d_in[4];
    const int*   pos_ids = (const int*)  d_in[5];
    const int*   k_quant = (const int*)  d_in[6];
    const float* k_scale = (const float*)d_in[7];
    const float* k_mn    = (const float*)d_in[8];
    const float* k_full  = (const float*)d_in[9];
    const float* key_p   = (const float*)d_in[10];
    const float* key_q   = (const float*)d_in[11];
    const int*   v_quant = (const int*)  d_in[12];
    const float* v_scale = (const float*)d_in[13];
    const float* v_mn    = (const float*)d_in[14];
    const float* v_full  = (const float*)d_in[15];
    const float* value_q = (const float*)d_in[16];
    const float* value_p = (const float*)d_in[17];

    float* ws   = (float*)d_ws;
    float* xpad = ws;                                   // (16, HID_)
    float* q_ws = xpad + MPAD_ * HID_;                  // (B_, HID_)
    float* k_ws = q_ws + B_ * HID_;
    float* v_ws = k_ws + B_ * HID_;
    float* apad = v_ws + B_ * HID_;                     // (16, HID_)

    dim3 blk(256);

    // Zero-pad inputs for WMMA GEMMs.
    init_pad_kernel<<<(MPAD_ * HID_) / 256, blk, 0, stream>>>(hidden, xpad, apad);

    // QKV projections (WMMA f32 16x16x4), 256 tiles / 8 waves per block.
    gemm16_kernel<<<32, blk, 0, stream>>>(xpad, wq, q_ws);
    gemm16_kernel<<<32, blk, 0, stream>>>(xpad, wk, k_ws);
    gemm16_kernel<<<32, blk, 0, stream>>>(xpad, wv, v_ws);

    // RoPE on q and k.
    rope_kernel<<<(2 * B_ * HID_) / 256, blk, 0, stream>>>(q_ws, k_ws, pos_ids);

    // Attention (one block per (b,h)); writes rows 0..3 of apad.
    attn_kernel<<<B_ * H_, blk, 0, stream>>>(
        q_ws, k_ws, v_ws,
        k_quant, k_scale, k_mn, k_full, key_p, key_q,
        v_quant, v_scale, v_mn, v_full, value_q, value_p,
        apad);

    // Output projection -> d_out.
    gemm16_kernel<<<32, blk, 0, stream>>>(apad, wo, (float*)d_out);
}